// RNNCell_34660386079254
// MI455X (gfx1250) — compile-verified
//
#include <hip/hip_runtime.h>
#include <math.h>

#define HS 512
#define IS 1024

typedef __attribute__((ext_vector_type(2))) float v2f;
typedef __attribute__((ext_vector_type(8))) float v8f;
typedef __attribute__((ext_vector_type(4))) float f4;

// ---------------------------------------------------------------------------
// Kernel A: a = W_ih @ x + W_hh @ h via fp32 WMMA (V_WMMA_F32_16X16X4_F32).
// One wave per 16-row tile (32 tiles total). B broadcasts the K-chunk across
// all 16 columns, so D[m, n] is the dot product for row m in every column n.
// ---------------------------------------------------------------------------
__global__ void __launch_bounds__(256) rnn_matvec_wmma(
    const float* __restrict__ W_ih, const float* __restrict__ x,
    const float* __restrict__ W_hh, const float* __restrict__ h,
    float* __restrict__ ws_a) {
  const int lane  = threadIdx.x & 31;
  const int wave  = threadIdx.x >> 5;
  const int tile  = blockIdx.x * 8 + wave;     // 0..31
  const int row0  = tile * 16;
  const int m     = lane & 15;                 // A: M = lane % 16
  const int khalf = (lane >> 4) * 2;           // K split {0,1} / {2,3}

  v8f acc = {0.f, 0.f, 0.f, 0.f, 0.f, 0.f, 0.f, 0.f};

  const float* __restrict__ Wr = W_ih + (size_t)(row0 + m) * IS;
  for (int k = 0; k < IS; k += 4) {
    v2f a, b;
    a.x = Wr[k + khalf];
    a.y = Wr[k + khalf + 1];
    b.x = x[k + khalf];
    b.y = x[k + khalf + 1];
    acc = __builtin_amdgcn_wmma_f32_16x16x4_f32(false, a, false, b,
                                                (short)0, acc, false, false);
  }

  const float* __restrict__ Wr2 = W_hh + (size_t)(row0 + m) * HS;
  for (int k = 0; k < HS; k += 4) {
    v2f a, b;
    a.x = Wr2[k + khalf];
    a.y = Wr2[k + khalf + 1];
    b.x = h[k + khalf];
    b.y = h[k + khalf + 1];
    acc = __builtin_amdgcn_wmma_f32_16x16x4_f32(false, a, false, b,
                                                (short)0, acc, false, false);
  }

  // C/D layout: lane L, VGPR v -> M = v + (L < 16 ? 0 : 8), value replicated
  // over all N columns. Lane 0 owns rows 0..7, lane 16 owns rows 8..15.
  if (lane == 0) {
#pragma unroll
    for (int v = 0; v < 8; ++v) ws_a[row0 + v] = acc[v];
  } else if (lane == 16) {
#pragma unroll
    for (int v = 0; v < 8; ++v) ws_a[row0 + 8 + v] = acc[v];
  }
}

// ---------------------------------------------------------------------------
// Kernel B: t = tanh(a + bias); new = heaviside(t, t); s = 1 - t^2
// ---------------------------------------------------------------------------
__global__ void rnn_activation(const float* __restrict__ ws_a,
                               const float* __restrict__ bias,
                               float* __restrict__ out_new,
                               float* __restrict__ ws_s) {
  int i = blockIdx.x * blockDim.x + threadIdx.x;   // 512 threads total
  float a = ws_a[i] + bias[i];
  float t = tanhf(a);
  out_new[i] = (t > 0.0f) ? 1.0f : (t == 0.0f ? t : 0.0f);
  ws_s[i] = 1.0f - t * t;
}

// ---------------------------------------------------------------------------
// Kernel C: streaming nontemporal zero-fill of bar_M (512 MB). The dominant
// cost of the whole problem -> B128 stores, NT hint to bypass L2 retention.
// ---------------------------------------------------------------------------
__global__ void __launch_bounds__(256) rnn_fill_zero(float* __restrict__ base,
                                                     size_t nvec4) {
  size_t idx    = (size_t)blockIdx.x * blockDim.x + threadIdx.x;
  size_t stride = (size_t)gridDim.x * blockDim.x;
  f4 z = {0.f, 0.f, 0.f, 0.f};
  for (size_t v = idx; v < nvec4; v += stride) {
    __builtin_nontemporal_store(z, (f4*)base + v);
  }
}

// ---------------------------------------------------------------------------
// Kernel D: J[i, j] = s[i] * W_hh[i, j]  (vectorized float4)
// ---------------------------------------------------------------------------
__global__ void rnn_jacobian(const float* __restrict__ W_hh,
                             const float* __restrict__ s,
                             float* __restrict__ J) {
  int idx = blockIdx.x * blockDim.x + threadIdx.x;  // 65536 vec4 = 512*512
  int e = idx * 4;
  int i = e >> 9;
  f4 w = *(const f4*)(W_hh + e);
  float si = s[i];
  f4 r = {w.x * si, w.y * si, w.z * si, w.w * si};
  *(f4*)(J + e) = r;
}

// ---------------------------------------------------------------------------
// Kernel E: diagonal scatter bar_M[i, i, k] = s[i] * h[k]  (runs after fill)
// ---------------------------------------------------------------------------
__global__ void rnn_diag(const float* __restrict__ s,
                         const float* __restrict__ h,
                         float* __restrict__ barM) {
  int i  = blockIdx.x;        // 512
  int kv = threadIdx.x;       // 128 threads -> float4 each
  float si = s[i];
  f4 hv = *(const f4*)(h + kv * 4);
  f4 r = {si * hv.x, si * hv.y, si * hv.z, si * hv.w};
  *(f4*)(barM + (size_t)i * (HS * HS) + (size_t)i * HS + kv * 4) = r;
}

extern "C" void kernel_launch(void* const* d_in, const int* in_sizes, int n_in,
                              void* d_out, int out_size, void* d_ws, size_t ws_size,
                              hipStream_t stream) {
  (void)in_sizes; (void)n_in; (void)out_size; (void)ws_size;

  const float* x    = (const float*)d_in[0];
  const float* h    = (const float*)d_in[1];
  const float* W_ih = (const float*)d_in[2];
  const float* W_hh = (const float*)d_in[3];
  const float* bias = (const float*)d_in[4];

  float* out     = (float*)d_out;
  float* out_new = out;                                   // 512
  float* barM    = out + HS;                              // 512^3
  float* Jout    = out + HS + (size_t)HS * HS * HS;       // 512^2

  float* ws_a = (float*)d_ws;     // 512 floats
  float* ws_s = ws_a + HS;        // 512 floats

  // 1) matvec via fp32 WMMA: 32 tiles = 4 blocks x 8 waves
  rnn_matvec_wmma<<<4, 256, 0, stream>>>(W_ih, x, W_hh, h, ws_a);
  // 2) activation: 512 threads
  rnn_activation<<<2, 256, 0, stream>>>(ws_a, bias, out_new, ws_s);
  // 3) 512 MB zero-fill of bar_M (the roofline cost): 2M threads, 16 B128 NT
  //    stores each
  rnn_fill_zero<<<8192, 256, 0, stream>>>(barM, (size_t)HS * HS * HS / 4);
  // 4) J = s[:,None] * W_hh
  rnn_jacobian<<<256, 256, 0, stream>>>(W_hh, ws_s, Jout);
  // 5) diagonal scatter into bar_M (after fill; stream-ordered)
  rnn_diag<<<512, 128, 0, stream>>>(ws_s, h, barM);
}